// GeneratedModel1_24472723653026
// MI455X (gfx1250) — compile-verified
//
#include <hip/hip_runtime.h>

// ---------------------------------------------------------------------------
// MI455X (gfx1250) implementation.
//
//   k_init    : zero stats + output accumulator
//   k_conv    : 7x7 s2 conv as implicit GEMM via V_WMMA_F32_16X16X4_F32
//               (LDS lookup table for im2col offsets -> no div/mod in loop)
//   k_stats   : per-channel sum / sumsq of x0 (b128 loads, LDS tree, atomics)
//   k_bnprep  : fold BN(mean,var) + gamma/beta into per-(branch,channel) A,B
//   k_fc      : fused 62-branch relu-sum (x1 lives only in LDS)
//               + split-K [16x100] GEMM via V_WMMA_F32_16X16X4_F32,
//               fc_w streamed non-temporally, x0 slab staged with
//               global_load_async_to_lds_b128 (ASYNCcnt path)
//   k_final   : add fc bias -> d_out
// ---------------------------------------------------------------------------

typedef float v2f __attribute__((ext_vector_type(2)));
typedef float v4f __attribute__((ext_vector_type(4)));
typedef float v8f __attribute__((ext_vector_type(8)));
typedef int   v4i __attribute__((vector_size(4 * sizeof(int))));

#define BN_EPS 1e-5f

#if __has_builtin(__builtin_amdgcn_global_load_async_to_lds_b128) && \
    __has_builtin(__builtin_amdgcn_s_wait_asynccnt)
#define HAVE_ASYNC_LDS 1
#endif

__device__ __forceinline__ v8f wmma4(v2f a, v2f b, v8f c) {
  return __builtin_amdgcn_wmma_f32_16x16x4_f32(
      false, a, false, b, (short)0, c, false, false);
}

// ---------------------------------------------------------------------------
__global__ __launch_bounds__(256) void k_init(float* ssum, float* ssq, float* accb) {
  int t = blockIdx.x * 256 + threadIdx.x;
  if (t < 64) { ssum[t] = 0.0f; ssq[t] = 0.0f; }
  if (t < 1600) accb[t] = 0.0f;
}

// ---------------------------------------------------------------------------
// Conv stem: x[16,3,224,224] -> x0[16,64,112,112], k7 s2 p3, bias fused.
// M = 128 pixels/workgroup (16 per wave), N = 64 oc (4 tiles), K = 147 -> 148.
__global__ __launch_bounds__(256) void k_conv(const float* __restrict__ x,
                                              const float* __restrict__ cw,
                                              const float* __restrict__ cb,
                                              float* __restrict__ x0) {
  __shared__ float wlds[64 * 149];            // [oc][K], stride 149 (bank-spread)
  __shared__ __align__(16) int tbl[150 * 2];  // {xoff, (kh<<16)|kw} per K index
  const int tid  = threadIdx.x;
  const int lane = tid & 31;
  const int wave = tid >> 5;
  const int b     = blockIdx.x / 98;
  const int ptile = blockIdx.x % 98;
  const int pixbase = ptile * 128 + wave * 16;

  for (int i = tid; i < 64 * 149; i += 256) {
    int oc = i / 149, kk = i % 149;
    wlds[i] = (kk < 147) ? cw[oc * 147 + kk] : 0.0f;
  }
  for (int i = tid; i < 150; i += 256) {
    int off = 0, khkw = (1 << 24);            // pad entry: kh huge -> always OOB
    if (i < 147) {
      int ic = i / 49, rr = i % 49;
      int kh = rr / 7, kw = rr % 7;
      off = ic * 50176 + kh * 224 + kw;
      khkw = (kh << 16) | kw;
    }
    tbl[i * 2]     = off;
    tbl[i * 2 + 1] = khkw;
  }
  __syncthreads();

  const int m   = lane & 15;                  // A-row (pixel) / B-col (oc)
  const int kof = (lane >> 4) << 1;           // K sub-offset: 0 or 2
  const int pix = pixbase + m;
  const int oh = pix / 112, ow = pix % 112;
  const int ihb = oh * 2 - 3, iwb = ow * 2 - 3;
  const int pbase = ihb * 224 + iwb;
  const float* xb = x + (size_t)b * (3 * 224 * 224);

  v8f acc[4] = {};
#pragma unroll 1   // keep accumulators stationary: no unroll-rotation mov storm
  for (int kb = 0; kb < 148; kb += 4) {
    // A fragment: two gathers via LDS offset table, branchless predication.
    int4 tt = *(const int4*)&tbl[(kb + kof) * 2];
    int kh0 = tt.y >> 16, kw0 = tt.y & 0xffff;
    int kh1 = tt.w >> 16, kw1 = tt.w & 0xffff;
    bool p0 = ((unsigned)(ihb + kh0) < 224u) & ((unsigned)(iwb + kw0) < 224u);
    bool p1 = ((unsigned)(ihb + kh1) < 224u) & ((unsigned)(iwb + kw1) < 224u);
    int i0 = p0 ? pbase + tt.x : 0;
    int i1 = p1 ? pbase + tt.z : 0;
    float f0 = xb[i0], f1 = xb[i1];
    v2f a;
    a[0] = p0 ? f0 : 0.0f;
    a[1] = p1 ? f1 : 0.0f;
#pragma unroll
    for (int t = 0; t < 4; ++t) {
      const float* wr = &wlds[(t * 16 + m) * 149 + kb + kof];
      v2f bf; bf[0] = wr[0]; bf[1] = wr[1];
      acc[t] = wmma4(a, bf, acc[t]);
    }
  }

  // C layout: VGPR r -> M=r (lanes 0-15) / M=8+r (lanes 16-31), N = lane&15.
  const int mhi = (lane >> 4) << 3;
#pragma unroll
  for (int t = 0; t < 4; ++t) {
    int oc = t * 16 + m;
    float bias = cb[oc];
    float* orow = x0 + (size_t)b * 802816 + (size_t)oc * 12544 + pixbase + mhi;
    v4f lo, hi;
#pragma unroll
    for (int r = 0; r < 4; ++r) { lo[r] = acc[t][r] + bias; hi[r] = acc[t][r + 4] + bias; }
    *(v4f*)(orow)     = lo;     // 16B-aligned: pixbase%16==0, mhi in {0,8}
    *(v4f*)(orow + 4) = hi;
  }
}

// ---------------------------------------------------------------------------
__global__ __launch_bounds__(256) void k_stats(const float* __restrict__ x0,
                                               float* __restrict__ ssum,
                                               float* __restrict__ ssq) {
  __shared__ float rs[256], rq[256];
  const int tid = threadIdx.x;
  const int b = blockIdx.x >> 6;
  const int c = blockIdx.x & 63;
  const float* p = x0 + (size_t)b * 802816 + (size_t)c * 12544;
  float s = 0.0f, q = 0.0f;
  for (int i = tid * 4; i < 12544; i += 1024) {
    v4f v = *(const v4f*)(p + i);
#pragma unroll
    for (int u = 0; u < 4; ++u) { s += v[u]; q += v[u] * v[u]; }
  }
  rs[tid] = s; rq[tid] = q;
  __syncthreads();
  for (int o = 128; o > 0; o >>= 1) {
    if (tid < o) { rs[tid] += rs[tid + o]; rq[tid] += rq[tid + o]; }
    __syncthreads();
  }
  if (tid == 0) { atomicAdd(&ssum[c], rs[0]); atomicAdd(&ssq[c], rq[0]); }
}

// ---------------------------------------------------------------------------
__global__ __launch_bounds__(256) void k_bnprep(const float* __restrict__ ssum,
                                                const float* __restrict__ ssq,
                                                const float* __restrict__ gamma,
                                                const float* __restrict__ beta,
                                                float* __restrict__ Atab,
                                                float* __restrict__ Btab) {
  __shared__ float mean_s[64], inv_s[64];
  const int tid = threadIdx.x;
  if (tid < 64) {
    float mn = ssum[tid] * (1.0f / 200704.0f);
    float vr = ssq[tid] * (1.0f / 200704.0f) - mn * mn;
    mean_s[tid] = mn;
    inv_s[tid]  = rsqrtf(vr + BN_EPS);
  }
  __syncthreads();
  for (int e = tid; e < 62 * 64; e += 256) {
    int c = e & 63;
    float A = gamma[e] * inv_s[c];
    Atab[e] = A;
    Btab[e] = beta[e] - A * mean_s[c];
  }
}

// ---------------------------------------------------------------------------
// Fused branch-sum + FC. One workgroup per K-slab of 896 flat elements
// (896*14 = 12544 -> slab lies in one channel). M=16 batches, N=100 (7 tiles
// of 16, last padded), split-K accumulated with f32 atomics.
__global__ __launch_bounds__(256) void k_fc(const float* __restrict__ x0,
                                            const float* __restrict__ Atab,
                                            const float* __restrict__ Btab,
                                            const float* __restrict__ fcw,
                                            float* __restrict__ accb) {
  __shared__ __align__(16) float xl[16 * 900];  // [batch][k], stride 900
  __shared__ float Al[62], Bl[62];
  const int tid = threadIdx.x, lane = tid & 31, wave = tid >> 5;
  const int slab = blockIdx.x;                  // 0..895
  const int k0 = slab * 896;
  const int c  = slab / 14;

#if defined(HAVE_ASYNC_LDS)
  // Stage x0 slab into LDS with the CDNA5 async-to-LDS path (ASYNCcnt).
  for (int e = tid * 4; e < 16 * 896; e += 256 * 4) {
    int b = e / 896, kk = e - b * 896;          // 16B chunks never cross rows
    const float* g = &x0[(size_t)b * 802816 + k0 + kk];
    __builtin_amdgcn_global_load_async_to_lds_b128(
        (v4i*)g, (v4i*)&xl[b * 900 + kk], 0, 0);
  }
#else
  for (int e = tid * 4; e < 16 * 896; e += 256 * 4) {
    int b = e / 896, kk = e - b * 896;
    *(v4f*)&xl[b * 900 + kk] = *(const v4f*)&x0[(size_t)b * 802816 + k0 + kk];
  }
#endif
  if (tid < 62) { Al[tid] = Atab[tid * 64 + c]; Bl[tid] = Btab[tid * 64 + c]; }
#if defined(HAVE_ASYNC_LDS)
  __builtin_amdgcn_s_wait_asynccnt(0);
#endif
  __syncthreads();

  // x1 = sum_i relu(A_i * x0 + B_i), computed in place in LDS.
  for (int e = tid; e < 16 * 896; e += 256) {
    int b = e / 896, kk = e - b * 896;
    float v = xl[b * 900 + kk];
    float r = 0.0f;
#pragma unroll 2
    for (int i = 0; i < 62; ++i) r += fmaxf(Al[i] * v + Bl[i], 0.0f);
    xl[b * 900 + kk] = r;
  }
  __syncthreads();

  // GEMM: out[16,100] += x1[16, slab] @ fcw[100, slab]^T
  const int wbase = wave * 112;                 // each wave owns 112 K values
  const int m   = lane & 15;                    // batch row (A) / oc col (B)
  const int kof = (lane >> 4) << 1;
  v8f acc[7] = {};

#pragma unroll
  for (int jt = 0; jt < 7; ++jt) {
    int j = jt * 16 + m;
    if (j < 100) __builtin_prefetch(&fcw[(size_t)j * 802816 + k0 + wbase], 0, 1);
  }

#pragma unroll 1   // keep the 56 accumulator VGPRs stationary across iterations
  for (int kb = 0; kb < 112; kb += 4) {
    v2f a = *(const v2f*)&xl[m * 900 + wbase + kb + kof];  // 8B-aligned
#pragma unroll
    for (int jt = 0; jt < 7; ++jt) {
      int j = jt * 16 + m;
      int jc = (j < 100) ? j : 99;                         // clamp, then mask
      const float* wr = &fcw[(size_t)jc * 802816 + k0 + wbase + kb + kof];
      v2f w2 = __builtin_nontemporal_load((const v2f*)wr); // stream-once: NT
      v2f bf;
      bf[0] = (j < 100) ? w2[0] : 0.0f;
      bf[1] = (j < 100) ? w2[1] : 0.0f;
      acc[jt] = wmma4(a, bf, acc[jt]);
    }
  }
  __syncthreads();                              // all waves done reading xl

  // Cross-wave reduction in LDS: part[wave][jt][r][lane]
  float* part = xl;                             // 8*7*8*32 = 14336 <= 14400
#pragma unroll
  for (int jt = 0; jt < 7; ++jt)
#pragma unroll
    for (int r = 0; r < 8; ++r)
      part[((wave * 7 + jt) * 8 + r) * 32 + lane] = acc[jt][r];
  __syncthreads();

  for (int t = tid; t < 1792; t += 256) {
    int jt = t >> 8;
    int rem = t & 255;
    int mm = rem >> 4, n = rem & 15;
    int r = mm & 7, ln = n + ((mm >> 3) << 4);
    float s = 0.0f;
#pragma unroll
    for (int w = 0; w < 8; ++w) s += part[((w * 7 + jt) * 8 + r) * 32 + ln];
    int j = jt * 16 + n;
    if (j < 100) atomicAdd(&accb[mm * 100 + j], s);
  }
}

// ---------------------------------------------------------------------------
__global__ __launch_bounds__(256) void k_final(const float* __restrict__ accb,
                                               const float* __restrict__ fcb,
                                               float* __restrict__ out) {
  int t = blockIdx.x * 256 + threadIdx.x;
  if (t < 1600) out[t] = accb[t] + fcb[t % 100];
}

// ---------------------------------------------------------------------------
extern "C" void kernel_launch(void* const* d_in, const int* in_sizes, int n_in,
                              void* d_out, int out_size, void* d_ws, size_t ws_size,
                              hipStream_t stream) {
  const float* x     = (const float*)d_in[0];
  const float* cw    = (const float*)d_in[1];
  const float* cb    = (const float*)d_in[2];
  const float* gamma = (const float*)d_in[3];
  const float* beta  = (const float*)d_in[4];
  const float* fcw   = (const float*)d_in[5];
  const float* fcb   = (const float*)d_in[6];
  float* out = (float*)d_out;

  float* ws   = (float*)d_ws;
  float* x0   = ws;                       // 12,845,056 floats (51.4 MB)
  float* ssum = ws + 12845056;            // 64
  float* ssq  = ssum + 64;                // 64
  float* Atab = ssq + 64;                 // 62*64
  float* Btab = Atab + 3968;              // 62*64
  float* accb = Btab + 3968;              // 1600

  k_init  <<<7,    256, 0, stream>>>(ssum, ssq, accb);
  k_conv  <<<1568, 256, 0, stream>>>(x, cw, cb, x0);
  k_stats <<<1024, 256, 0, stream>>>(x0, ssum, ssq);
  k_bnprep<<<1,    256, 0, stream>>>(ssum, ssq, gamma, beta, Atab, Btab);
  k_fc    <<<896,  256, 0, stream>>>(x0, Atab, Btab, fcw, accb);
  k_final <<<7,    256, 0, stream>>>(accb, fcb, out);
}